// StreamingDPM_22797686407983
// MI455X (gfx1250) — compile-verified
//
#include <hip/hip_runtime.h>
#include <math.h>

typedef __attribute__((ext_vector_type(2))) float v2f;
typedef __attribute__((ext_vector_type(8))) float v8f;

constexpr int N = 4096;
constexpr int D = 256;
constexpr int K = 64;
constexpr int ROW_TILES = N / 16;        // 256
constexpr int LDS_STRIDE = D + 4;        // 260 floats: lane l hits banks 4l..4l+1
constexpr int LDS_PER_WAVE = 16 * LDS_STRIDE;  // floats

// ---------------------------------------------------------------------------
// Kernel 1: per-cluster Cholesky (in LDS), logdet, and W = L^{-1} to global.
// One 256-thread block per cluster; dynamic LDS = D*D floats (256 KB).
// ---------------------------------------------------------------------------
__global__ void chol_inv_kernel(const float* __restrict__ covs,
                                float* __restrict__ W,
                                float* __restrict__ logdet) {
  extern __shared__ float S[];                 // D*D floats
  const int k   = blockIdx.x;
  const int tid = threadIdx.x;
  const float* A = covs + (size_t)k * D * D;

  for (int idx = tid; idx < D * D; idx += (int)blockDim.x) S[idx] = A[idx];
  __syncthreads();

  // Left-looking Cholesky: lower triangle of S becomes L.
  for (int j = 0; j < D; ++j) {
    if (tid == 0) {
      float s = S[j * D + j];
      for (int p = 0; p < j; ++p) { float l = S[j * D + p]; s -= l * l; }
      S[j * D + j] = sqrtf(fmaxf(s, 1e-30f));
    }
    __syncthreads();
    const float inv_djj = 1.0f / S[j * D + j];
    for (int i = j + 1 + tid; i < D; i += (int)blockDim.x) {
      float s = S[i * D + j];
      for (int p = 0; p < j; ++p) s -= S[i * D + p] * S[j * D + p];
      S[i * D + j] = s * inv_djj;
    }
    __syncthreads();
  }

  if (tid == 0) {
    float ld = 0.0f;
    for (int j = 0; j < D; ++j) ld += logf(S[j * D + j]);
    logdet[k] = 2.0f * ld;                     // log det(Sigma)
  }

  // W = L^{-1}: thread c solves L w = e_c by forward substitution.
  {
    const int c = tid;                         // blockDim.x == D
    float w[D];
    w[c] = 1.0f / S[c * D + c];
    for (int i = c + 1; i < D; ++i) {
      float s = 0.0f;
      for (int p = c; p < i; ++p) s += S[i * D + p] * w[p];
      w[i] = -s / S[i * D + i];
    }
    float* Wk = W + (size_t)k * D * D;
    for (int i = 0; i < c; ++i) Wk[i * D + c] = 0.0f;   // upper triangle = 0
    for (int i = c; i < D; ++i) Wk[i * D + c] = w[i];
  }
}

// ---------------------------------------------------------------------------
// Kernel 2: quad[n,k] = || W_k (x_n - m_k) ||^2 via V_WMMA_F32_16X16X4_F32.
// One wave owns one (16-row tile, cluster) pair.  The dev tile is staged in
// LDS once per wave (reused by all 16 Z column tiles); the main loop computes
// 4 column tiles per pass so each K-chunk costs 1 ds_load + 4 global loads
// for 4 WMMAs (1 vmem per wmma).
// ---------------------------------------------------------------------------
__global__ void quad_kernel(const float* __restrict__ x,
                            const float* __restrict__ means,
                            const float* __restrict__ W,
                            float* __restrict__ quad) {
  extern __shared__ float dev_s[];                 // 8 waves * 16 * LDS_STRIDE
  const int lane = threadIdx.x & 31;
  const int wave = threadIdx.x >> 5;
  const int widx = blockIdx.x * 8 + wave;          // pair index
  const int k    = widx >> 8;                      // widx / ROW_TILES
  const int rt   = widx & (ROW_TILES - 1);
  const int row0 = rt * 16;

  float* dv = dev_s + wave * LDS_PER_WAVE;

  // ---- stage dev = x - mean for this 16-row tile into LDS (per wave) ----
  {
    const float* xt = x + (size_t)row0 * D;
    const float* mk = means + (size_t)k * D;
    for (int idx = lane; idx < 16 * D; idx += 32) {
      const int r = idx >> 8;                      // row in tile
      const int c = idx & (D - 1);                 // column
      dv[r * LDS_STRIDE + c] = xt[(size_t)r * D + c] - mk[c];
    }
  }
  // LDS ops of one wave execute in order: own-wave read-after-write is safe.

  // ISA 32-bit A 16x4 layout: lanes 0-15 -> K=0,1 ; lanes 16-31 -> K=2,3.
  const int half = lane >> 4;                      // 0 or 1
  const int l    = lane & 15;                      // M (A) / N (B) index
  const int koff = half * 2;

  const float* Wk = W + (size_t)k * D * D;
  const float* arow = dv + l * LDS_STRIDE + koff;  // A fragment base (LDS)

  float acc[8] = {0.f, 0.f, 0.f, 0.f, 0.f, 0.f, 0.f, 0.f};

  for (int cb = 0; cb < D / 64; ++cb) {            // 4 blocks of 4 col tiles
    const float* w0 = Wk + (size_t)(cb * 64 +  0 + l) * D + koff;
    const float* w1 = Wk + (size_t)(cb * 64 + 16 + l) * D + koff;
    const float* w2 = Wk + (size_t)(cb * 64 + 32 + l) * D + koff;
    const float* w3 = Wk + (size_t)(cb * 64 + 48 + l) * D + koff;
    v8f c0 = {}, c1 = {}, c2 = {}, c3 = {};
#pragma unroll 2
    for (int kk = 0; kk < D; kk += 4) {            // 64 WMMA quads per block
      v2f a  = *(const v2f*)(arow + kk);           // shared A fragment (LDS)
      v2f b0 = *(const v2f*)(w0 + kk);
      v2f b1 = *(const v2f*)(w1 + kk);
      v2f b2 = *(const v2f*)(w2 + kk);
      v2f b3 = *(const v2f*)(w3 + kk);
      c0 = __builtin_amdgcn_wmma_f32_16x16x4_f32(false, a, false, b0,
                                                 (short)0, c0, false, false);
      c1 = __builtin_amdgcn_wmma_f32_16x16x4_f32(false, a, false, b1,
                                                 (short)0, c1, false, false);
      c2 = __builtin_amdgcn_wmma_f32_16x16x4_f32(false, a, false, b2,
                                                 (short)0, c2, false, false);
      c3 = __builtin_amdgcn_wmma_f32_16x16x4_f32(false, a, false, b3,
                                                 (short)0, c3, false, false);
    }
#pragma unroll
    for (int r = 0; r < 8; ++r)
      acc[r] += c0[r] * c0[r] + c1[r] * c1[r] + c2[r] * c2[r] + c3[r] * c3[r];
  }

  // C/D layout: lanes 0-15 hold rows 0-7, lanes 16-31 hold rows 8-15;
  // columns stripe across the 16 lanes of each half -> butterfly within half.
#pragma unroll
  for (int r = 0; r < 8; ++r) {
    float v = acc[r];
    v += __shfl_xor(v, 1, 32);
    v += __shfl_xor(v, 2, 32);
    v += __shfl_xor(v, 4, 32);
    v += __shfl_xor(v, 8, 32);
    acc[r] = v;
  }
  if (l == 0) {
#pragma unroll
    for (int r = 0; r < 8; ++r)
      quad[(size_t)(row0 + half * 8 + r) * K + k] = acc[r];
  }
}

// ---------------------------------------------------------------------------
// Kernel 3: row softmax over [log_lik + log_prior | log_new].  One wave/row.
// ---------------------------------------------------------------------------
__global__ void softmax_kernel(const float* __restrict__ quad,
                               const float* __restrict__ logdet,
                               const float* __restrict__ counts,
                               const int* __restrict__ n_points,
                               float* __restrict__ out) {
  const int lane = threadIdx.x & 31;
  const int row  = blockIdx.x * 8 + (threadIdx.x >> 5);
  if (row >= N) return;

  const float denom = (float)(*n_points) + 1.0f;                 // ALPHA = 1
  const float cterm = -0.5f * (float)D * 1.8378770664093453f;    // -D/2*log(2pi)

  float lp[3];
#pragma unroll
  for (int i = 0; i < 3; ++i) {
    const int kk = lane + 32 * i;
    if (kk < K) {
      lp[i] = cterm - 0.5f * (logdet[kk] + quad[(size_t)row * K + kk])
              + logf(counts[kk] / denom);
    } else if (kk == K) {
      lp[i] = logf(1.0f / denom);                                // log_new
    } else {
      lp[i] = -3.0e38f;
    }
  }

  float m = fmaxf(fmaxf(lp[0], lp[1]), lp[2]);
  m = fmaxf(m, __shfl_xor(m, 1, 32));
  m = fmaxf(m, __shfl_xor(m, 2, 32));
  m = fmaxf(m, __shfl_xor(m, 4, 32));
  m = fmaxf(m, __shfl_xor(m, 8, 32));
  m = fmaxf(m, __shfl_xor(m, 16, 32));

  float e[3];
  float s = 0.0f;
#pragma unroll
  for (int i = 0; i < 3; ++i) {
    const int kk = lane + 32 * i;
    e[i] = (kk <= K) ? expf(lp[i] - m) : 0.0f;
    s += e[i];
  }
  s += __shfl_xor(s, 1, 32);
  s += __shfl_xor(s, 2, 32);
  s += __shfl_xor(s, 4, 32);
  s += __shfl_xor(s, 8, 32);
  s += __shfl_xor(s, 16, 32);
  const float inv_s = 1.0f / s;

#pragma unroll
  for (int i = 0; i < 3; ++i) {
    const int kk = lane + 32 * i;
    if (kk <= K) out[(size_t)row * (K + 1) + kk] = e[i] * inv_s;
  }
}

// ---------------------------------------------------------------------------
extern "C" void kernel_launch(void* const* d_in, const int* in_sizes, int n_in,
                              void* d_out, int out_size, void* d_ws, size_t ws_size,
                              hipStream_t stream) {
  const float* x        = (const float*)d_in[0];   // [N, D]
  const float* means    = (const float*)d_in[1];   // [K, D]
  const float* covs     = (const float*)d_in[2];   // [K, D, D]
  const float* counts   = (const float*)d_in[3];   // [K]
  const int*   n_points = (const int*)d_in[4];     // scalar
  float*       out      = (float*)d_out;           // [N, K+1]

  char* ws = (char*)d_ws;
  float* W      = (float*)ws;                                  // K*D*D floats
  float* logdet = (float*)(ws + (size_t)K * D * D * sizeof(float));   // K floats
  float* quad   = (float*)(ws + (size_t)K * D * D * sizeof(float) + 1024); // N*K

  // 1) Cholesky + inverse + logdet: one block per cluster, 256 KB dynamic LDS.
  chol_inv_kernel<<<K, D, D * D * sizeof(float), stream>>>(covs, W, logdet);

  // 2) Mahalanobis quadratic forms via fp32 WMMA with LDS-staged dev tiles.
  quad_kernel<<<(K * ROW_TILES) / 8, 256,
                8 * LDS_PER_WAVE * sizeof(float), stream>>>(x, means, W, quad);

  // 3) Row softmax over K+1 entries: one wave per row.
  softmax_kernel<<<N / 8, 256, 0, stream>>>(quad, logdet, counts, n_points, out);
}